// GCNLayer_23347442221165
// MI455X (gfx1250) — compile-verified
//
#include <hip/hip_runtime.h>
#include <math.h>

// Problem constants from the reference
#define BATCH 2048
#define NN_   118     // nodes
#define NP_   120     // nodes padded to multiple of 4 (WMMA K step)
#define DD_   64      // feature dim
#define OUTD  64      // output dim

typedef float v2f __attribute__((ext_vector_type(2)));
typedef float v4f __attribute__((ext_vector_type(4)));
typedef float v8f __attribute__((ext_vector_type(8)));

// D = A(16x4 f32) * B(4x16 f32) + C(16x16 f32)
__device__ __forceinline__ v8f wmma4(v2f a, v2f b, v8f c) {
  return __builtin_amdgcn_wmma_f32_16x16x4_f32(
      /*neg_a=*/false, a, /*neg_b=*/false, b,
      /*c_mod=*/(short)0, c, /*reuse_a=*/false, /*reuse_b=*/false);
}

__device__ __forceinline__ float sigmoidf_(float x) {
  return 1.0f / (1.0f + __expf(-x));
}

// Branch-free masked A-fragment load: 8B vector load from a clamped
// (always-in-bounds) address, then cndmask the two lanes' values.
__device__ __forceinline__ v2f ldA(const float* __restrict__ base,
                                   int rowSafe, int kkSafe, bool valid) {
  v2f raw = __builtin_nontemporal_load(
      (const v2f*)(base + (size_t)rowSafe * NN_ + kkSafe));
  v2f r;
  r.x = valid ? raw.x : 0.0f;
  r.y = valid ? raw.y : 0.0f;
  return r;
}

__launch_bounds__(256, 1)
__global__ void pf_gcn_fused(
    const float* __restrict__ e,   const float* __restrict__ f,
    const float* __restrict__ k1,  const float* __restrict__ k2,
    const float* __restrict__ Gnd, const float* __restrict__ Bnd,
    const float* __restrict__ Gdg, const float* __restrict__ Bdg,
    const float* __restrict__ Pd,  const float* __restrict__ Qd,
    const float* __restrict__ W1,  const float* __restrict__ b1,
    const float* __restrict__ W2,  const float* __restrict__ b2,
    const float* __restrict__ wae, const float* __restrict__ bae,
    const float* __restrict__ waf, const float* __restrict__ baf,
    float* __restrict__ out)
{
  // --- LDS tiles: 10 x (120x64 f32) = 300 KB + small scalars ---
  __shared__ float sE [NP_*DD_], sF [NP_*DD_];
  __shared__ float sE3[NP_*DD_], sF3[NP_*DD_];
  __shared__ float sNE[NP_*DD_], sNF[NP_*DD_];
  __shared__ float sE1[NP_*DD_], sF1[NP_*DD_];
  __shared__ float sE2[NP_*DD_], sF2[NP_*DD_];
  __shared__ float sPd[128], sQd[128], sGd[128], sBd[128];
  __shared__ float sWae[DD_], sWaf[DD_];
  __shared__ float sSum[8];
  __shared__ float sCe[5], sCf[5];

  const int b    = blockIdx.x;
  const int tid  = threadIdx.x;
  const int w    = tid >> 5;          // wave id: row strip 16w..16w+15
  const int lane = tid & 31;
  const int l16  = lane & 15;
  const int koff = (lane < 16) ? 0 : 2;          // A/B fragment K sub-offset
  const int rowA = w * 16 + l16;                 // A-fragment row (per lane)
  const int rowC = w * 16 + ((lane >> 4) << 3);  // C/D base row (add r 0..7)
  const bool rOK = (rowA < NN_);
  const int rowSafe = rOK ? rowA : 0;

  // ---------------- stage 0: cooperative loads ----------------
  const int ND = NN_ * DD_;          // 7552, multiple of 4
  const float* eB = e + (size_t)b * ND;
  const float* fB = f + (size_t)b * ND;
  for (int i = tid * 4; i < NP_ * DD_; i += 256 * 4) {
    v4f ve = (v4f)(0.0f), vf = (v4f)(0.0f);
    if (i < ND) {
      ve = __builtin_nontemporal_load((const v4f*)(eB + i));
      vf = __builtin_nontemporal_load((const v4f*)(fB + i));
    }
    *(v4f*)(sE + i) = ve;
    *(v4f*)(sF + i) = vf;
  }
  if (tid < 128) {
    if (tid < NN_) {
      sPd[tid] = Pd [(size_t)b * NN_ + tid];
      sQd[tid] = Qd [(size_t)b * NN_ + tid];
      sGd[tid] = Gdg[(size_t)b * NN_ + tid];
      sBd[tid] = Bdg[(size_t)b * NN_ + tid];
    } else {
      sPd[tid] = 0.0f; sQd[tid] = 0.0f; sGd[tid] = 1.0f; sBd[tid] = 1.0f;
    }
  }
  if (tid < DD_) { sWae[tid] = wae[tid]; sWaf[tid] = waf[tid]; }
  if (tid < 8)   sSum[tid] = 0.0f;
  __syncthreads();

  const size_t NNsz = (size_t)NN_ * NN_;
  const float* Gb  = Gnd + (size_t)b * NNsz;
  const float* Bb  = Bnd + (size_t)b * NNsz;
  const float* K1b = k1  + (size_t)b * NNsz;
  const float* K2b = k2  + (size_t)b * NNsz;

  // ============ stage 1: Ge, Gf, Be, Bf via f32 WMMA ============
  v8f aGe[4], aGf[4], aBe[4], aBf[4];
  #pragma unroll
  for (int t = 0; t < 4; ++t) {
    aGe[t] = (v8f)(0.0f); aGf[t] = (v8f)(0.0f);
    aBe[t] = (v8f)(0.0f); aBf[t] = (v8f)(0.0f);
  }
  #pragma unroll 2
  for (int k0 = 0; k0 < NP_; k0 += 4) {
    const int  kk     = k0 + koff;           // even; invalid only at 118
    const bool kOK    = (kk + 1 < NN_);
    const int  kkSafe = kOK ? kk : 0;
    const bool valid  = rOK && kOK;
    const v2f aG = ldA(Gb, rowSafe, kkSafe, valid);
    const v2f aB = ldA(Bb, rowSafe, kkSafe, valid);
    #pragma unroll
    for (int t = 0; t < 4; ++t) {
      const int c0 = t * 16 + l16;
      v2f be, bf;
      be.x = sE[ kk      * DD_ + c0]; be.y = sE[(kk + 1) * DD_ + c0];
      bf.x = sF[ kk      * DD_ + c0]; bf.y = sF[(kk + 1) * DD_ + c0];
      aGe[t] = wmma4(aG, be, aGe[t]);
      aGf[t] = wmma4(aG, bf, aGf[t]);
      aBe[t] = wmma4(aB, be, aBe[t]);
      aBf[t] = wmma4(aB, bf, aBf[t]);
    }
  }

  // power-flow elementwise math on the C-layout tiles (all in registers)
  float pe3 = 0.0f, pne = 0.0f, pf3 = 0.0f, pnf = 0.0f;
  #pragma unroll
  for (int t = 0; t < 4; ++t) {
    #pragma unroll
    for (int r = 0; r < 8; ++r) {
      const int row = rowC + r;
      const int col = t * 16 + l16;
      const float ge = aGe[t][r], gf = aGf[t][r];
      const float be = aBe[t][r], bf = aBf[t][r];
      const int safeRow = (row < NP_) ? row : 0;   // pads handled by guards
      const float ev = sE[safeRow * DD_ + col];
      const float fv = sF[safeRow * DD_ + col];
      const float pd = sPd[row & 127], qd = sQd[row & 127];
      const float gd = sGd[row & 127], bd = sBd[row & 127];
      const float ef2  = ev * ev + fv * fv;
      const float ib   = 1.0f / (ef2 + 0.1f);
      const float alpha = (pd * ev + qd * fv) * ib - ge - bf;
      const float beta  = (qd * ev - pd * fv) * ib + gf + bf; // faithful: Bf
      const float igb  = 1.0f / (gd * gd + bd * bd);
      const float e3v  = (alpha * gd + beta * bd) * igb;
      const float f3v  = (beta * gd - alpha * bd) * igb;
      const float base1 = ge - bf;
      const float base2 = gf + be;
      const float Ptv = pd - ef2 * gd;
      const float Qtv = qd + ef2 * bd;
      const float nev = (Ptv * base1 + Qtv * base2) * igb;
      const float nfv = (Ptv * base2 - Qtv * base1) * igb;
      if (row < NP_) {
        sE3[row * DD_ + col] = e3v;  sF3[row * DD_ + col] = f3v;
        sNE[row * DD_ + col] = nev;  sNF[row * DD_ + col] = nfv;
      }
      if (row < NN_) {
        const float wa = sWae[col], wb = sWaf[col];
        pe3 += e3v * wa;  pne += nev * wa;
        pf3 += f3v * wb;  pnf += nfv * wb;
      }
    }
  }
  #pragma unroll
  for (int off = 16; off > 0; off >>= 1) {
    pe3 += __shfl_xor(pe3, off, 32);
    pne += __shfl_xor(pne, off, 32);
    pf3 += __shfl_xor(pf3, off, 32);
    pnf += __shfl_xor(pnf, off, 32);
  }
  if (lane == 0) {
    atomicAdd(&sSum[0], pe3); atomicAdd(&sSum[1], pne);
    atomicAdd(&sSum[4], pf3); atomicAdd(&sSum[5], pnf);
  }

  // ============ stage 2: e1,f1,e2,f2 = k1/k2 @ e,f ============
  v8f aE1[4], aF1[4], aE2[4], aF2[4];
  #pragma unroll
  for (int t = 0; t < 4; ++t) {
    aE1[t] = (v8f)(0.0f); aF1[t] = (v8f)(0.0f);
    aE2[t] = (v8f)(0.0f); aF2[t] = (v8f)(0.0f);
  }
  #pragma unroll 2
  for (int k0 = 0; k0 < NP_; k0 += 4) {
    const int  kk     = k0 + koff;
    const bool kOK    = (kk + 1 < NN_);
    const int  kkSafe = kOK ? kk : 0;
    const bool valid  = rOK && kOK;
    const v2f a1v = ldA(K1b, rowSafe, kkSafe, valid);
    const v2f a2v = ldA(K2b, rowSafe, kkSafe, valid);
    #pragma unroll
    for (int t = 0; t < 4; ++t) {
      const int c0 = t * 16 + l16;
      v2f be, bf;
      be.x = sE[ kk      * DD_ + c0]; be.y = sE[(kk + 1) * DD_ + c0];
      bf.x = sF[ kk      * DD_ + c0]; bf.y = sF[(kk + 1) * DD_ + c0];
      aE1[t] = wmma4(a1v, be, aE1[t]);
      aF1[t] = wmma4(a1v, bf, aF1[t]);
      aE2[t] = wmma4(a2v, be, aE2[t]);
      aF2[t] = wmma4(a2v, bf, aF2[t]);
    }
  }
  float p1 = 0.0f, p2 = 0.0f, q1 = 0.0f, q2 = 0.0f;
  #pragma unroll
  for (int t = 0; t < 4; ++t) {
    #pragma unroll
    for (int r = 0; r < 8; ++r) {
      const int row = rowC + r;
      const int col = t * 16 + l16;
      if (row < NP_) {
        sE1[row * DD_ + col] = aE1[t][r];  sF1[row * DD_ + col] = aF1[t][r];
        sE2[row * DD_ + col] = aE2[t][r];  sF2[row * DD_ + col] = aF2[t][r];
      }
      if (row < NN_) {
        const float wa = sWae[col], wb = sWaf[col];
        p1 += aE1[t][r] * wa;  p2 += aE2[t][r] * wa;
        q1 += aF1[t][r] * wb;  q2 += aF2[t][r] * wb;
      }
    }
  }
  #pragma unroll
  for (int off = 16; off > 0; off >>= 1) {
    p1 += __shfl_xor(p1, off, 32);
    p2 += __shfl_xor(p2, off, 32);
    q1 += __shfl_xor(q1, off, 32);
    q2 += __shfl_xor(q2, off, 32);
  }
  if (lane == 0) {
    atomicAdd(&sSum[2], p1); atomicAdd(&sSum[3], p2);
    atomicAdd(&sSum[6], q1); atomicAdd(&sSum[7], q2);
  }
  __syncthreads();

  // ============ attention coefficients (one thread) ============
  if (tid == 0) {
    const float inv = 1.0f / (float)NN_;
    const float ba = bae[0], bb = baf[0];
    const float a1 = sigmoidf_(sSum[0] * inv + ba);
    const float a2 = sigmoidf_(sSum[1] * inv + ba);
    const float a3 = sigmoidf_(sSum[2] * inv + ba);
    const float a4 = sigmoidf_(sSum[3] * inv + ba);
    const float ab = a1 + a2 + a3 + a4 + 1e-4f;
    sCe[0] = a1 / ab; sCe[1] = a2 / ab; sCe[2] = a3 / ab; sCe[3] = a4 / ab;
    sCe[4] = 1.0f;
    const float c1 = sigmoidf_(sSum[4] * inv + bb);
    const float c2 = sigmoidf_(sSum[5] * inv + bb);
    const float c3 = sigmoidf_(sSum[6] * inv + bb);
    const float c4 = sigmoidf_(sSum[7] * inv + bb);
    const float cb = c1 + c2 + c3 + c4 + 1e-4f;
    sCf[0] = c1 / cb; sCf[1] = c2 / cb; sCf[2] = c3 / cb; sCf[3] = c4 / cb;
    sCf[4] = 1.0f;
  }
  __syncthreads();

  // ============ stage 3: fused output GEMMs (concat folded) ============
  // e_out = relu( sum_i c_i * (X_i @ W1_blk_i) + b1 ), same for f side.
  v8f o1[4], o2[4];
  #pragma unroll
  for (int t = 0; t < 4; ++t) { o1[t] = (v8f)(0.0f); o2[t] = (v8f)(0.0f); }

  const float* eSrc[5] = { sE3, sNE, sE1, sE2, sE };
  const float* fSrc[5] = { sF3, sNF, sF1, sF2, sF };

  #pragma unroll
  for (int term = 0; term < 5; ++term) {
    const float* xe = eSrc[term];
    const float* xf = fSrc[term];
    const float ce = sCe[term];
    const float cf = sCf[term];
    const float* w1t = W1 + (size_t)(term * DD_) * OUTD;
    const float* w2t = W2 + (size_t)(term * DD_) * OUTD;
    #pragma unroll 4
    for (int k0 = 0; k0 < DD_; k0 += 4) {
      const int kk = k0 + koff;
      // A fragments: one 8B LDS load each (even offset), scaled by c_i
      v2f xre = *(const v2f*)(xe + rowA * DD_ + kk);
      v2f xrf = *(const v2f*)(xf + rowA * DD_ + kk);
      v2f ae, af;
      ae.x = ce * xre.x; ae.y = ce * xre.y;
      af.x = cf * xrf.x; af.y = cf * xrf.y;
      #pragma unroll
      for (int t = 0; t < 4; ++t) {
        const int c0 = t * 16 + l16;
        v2f b1v, b2v;
        b1v.x = w1t[ kk      * OUTD + c0]; b1v.y = w1t[(kk + 1) * OUTD + c0];
        b2v.x = w2t[ kk      * OUTD + c0]; b2v.y = w2t[(kk + 1) * OUTD + c0];
        o1[t] = wmma4(ae, b1v, o1[t]);
        o2[t] = wmma4(af, b2v, o2[t]);
      }
    }
  }

  // bias + relu + store (tuple: e_out flat, then f_out flat)
  const size_t outE = (size_t)b * ND;
  const size_t outF = (size_t)BATCH * ND + (size_t)b * ND;
  #pragma unroll
  for (int t = 0; t < 4; ++t) {
    #pragma unroll
    for (int r = 0; r < 8; ++r) {
      const int row = rowC + r;
      if (row < NN_) {
        const int col = t * 16 + l16;
        const float v1 = fmaxf(o1[t][r] + b1[col], 0.0f);
        const float v2 = fmaxf(o2[t][r] + b2[col], 0.0f);
        __builtin_nontemporal_store(v1, out + outE + row * DD_ + col);
        __builtin_nontemporal_store(v2, out + outF + row * DD_ + col);
      }
    }
  }
}

extern "C" void kernel_launch(void* const* d_in, const int* in_sizes, int n_in,
                              void* d_out, int out_size, void* d_ws, size_t ws_size,
                              hipStream_t stream) {
  (void)in_sizes; (void)n_in; (void)out_size; (void)d_ws; (void)ws_size;
  const float* e   = (const float*)d_in[0];
  const float* f   = (const float*)d_in[1];
  const float* k1  = (const float*)d_in[2];
  const float* k2  = (const float*)d_in[3];
  const float* Gnd = (const float*)d_in[4];
  const float* Bnd = (const float*)d_in[5];
  const float* Gdg = (const float*)d_in[6];
  const float* Bdg = (const float*)d_in[7];
  const float* Pd  = (const float*)d_in[8];
  const float* Qd  = (const float*)d_in[9];
  const float* W1  = (const float*)d_in[10];
  const float* b1  = (const float*)d_in[11];
  const float* W2  = (const float*)d_in[12];
  const float* b2  = (const float*)d_in[13];
  const float* wae = (const float*)d_in[14];
  const float* bae = (const float*)d_in[15];
  const float* waf = (const float*)d_in[16];
  const float* baf = (const float*)d_in[17];
  float* out = (float*)d_out;

  pf_gcn_fused<<<BATCH, 256, 0, stream>>>(
      e, f, k1, k2, Gnd, Bnd, Gdg, Bdg, Pd, Qd,
      W1, b1, W2, b2, wae, bae, waf, baf, out);
}